// VolumetricCycleConsistencyLoss_43310450213285
// MI455X (gfx1250) — compile-verified
//
#include <hip/hip_runtime.h>

// ---------------------------------------------------------------------------
// Multi-scale SSIM + L1 loss, fused per-scale kernel for MI455X (gfx1250).
//
// The separable 11-tap Gaussian blur is cast as 16x16 = (16xK)x(Kx16) banded
// matmuls and run on V_WMMA_F32_16X16X4_F32 (full fp32: sigma = E[x^2]-mu^2
// is cancellation-sensitive, so fp16/fp8 WMMA flavors are rejected).
// One wave32 per 16x16 (H,W) tile streams depth with an 11-plane LDS ring.
// ---------------------------------------------------------------------------

#define TS    16          // output tile edge (H,W)
#define HALO  5
#define INW   26          // TS + 2*HALO
#define INP   28          // padded to multiple of 4 (WMMA K-chunks)
#define WIN   11
#define NF    5           // x, y, xx, yy, xy
#define PLANE 256         // TS*TS

typedef float v2f __attribute__((ext_vector_type(2)));
typedef float v8f __attribute__((ext_vector_type(8)));

// Normalized gaussian, sigma = 1.5, window 11.
__constant__ float GW[11] = {
    0.0010284f, 0.0075988f, 0.0360007f, 0.1093607f, 0.2130055f,
    0.2660118f,
    0.2130055f, 0.1093607f, 0.0360007f, 0.0075988f, 0.0010284f};

__device__ __forceinline__ float gband(int k, int n) {
  int d = k - n;
  return (d >= 0 && d <= 10) ? GW[d] : 0.0f;
}

__device__ __forceinline__ v8f wmma4(v2f a, v2f b, v8f c) {
  // D = A(16x4,f32) * B(4x16,f32) + C(16x16,f32)
  return __builtin_amdgcn_wmma_f32_16x16x4_f32(
      /*neg_a=*/false, a, /*neg_b=*/false, b,
      /*c_mod=*/(short)0, c, /*reuse_a=*/false, /*reuse_b=*/false);
}

// ---------------------------------------------------------------------------
// Fused L1 + SSIM partial-sums kernel. blockDim = 32 (one wave per tile).
// grid.x = 2 * (S/16)^2   (batch * HW tiles)
// ---------------------------------------------------------------------------
__global__ void msssim_scale_kernel(const float* __restrict__ X,
                                    const float* __restrict__ Y,
                                    int S,
                                    float* __restrict__ pl1,
                                    float* __restrict__ pss) {
  extern __shared__ float smem[];
  float* xpl  = smem;                        // INW * INP
  float* ypl  = xpl + INW * INP;             // INW * INP
  float* midT = ypl + INW * INP;             // NF * 16 * INP   (col-major rows)
  float* ring = midT + NF * 16 * INP;        // NF * WIN * PLANE

  const int lane = threadIdx.x;              // 0..31
  const int m    = lane & 15;
  const int hi   = lane >> 4;

  const int tilesPerRow = S / TS;
  const int tilesPerVol = tilesPerRow * tilesPerRow;
  int tid = blockIdx.x;
  int b   = tid / tilesPerVol;
  int t   = tid - b * tilesPerVol;
  int th  = t / tilesPerRow;
  int h0  = th * TS;
  int w0  = (t - th * tilesPerRow) * TS;

  const long long volOff = (long long)b * S * S * S;
  const float* xb = X + volOff;
  const float* yb = Y + volOff;

  // Banded gaussian operands, identical per-lane value for stage-1 B (G[k][n],
  // n = lane&15) and stage-2 A (G^T[m][k] = G[k][m], m = lane&15).
  v2f gw[7];
#pragma unroll
  for (int j = 0; j < 7; ++j) {
    int klo = 4 * j + (hi << 1);
    gw[j].x = gband(klo, m);
    gw[j].y = gband(klo + 1, m);
  }

  // Zero ring buffer (implements zero 'SAME' padding in D) and padded tails.
  for (int i = lane; i < NF * WIN * PLANE; i += 32) ring[i] = 0.0f;
  for (int r = lane; r < INW; r += 32) {
    xpl[r * INP + 26] = 0.0f; xpl[r * INP + 27] = 0.0f;
    ypl[r * INP + 26] = 0.0f; ypl[r * INP + 27] = 0.0f;
  }
  for (int i = lane; i < NF * 16; i += 32) {
    midT[i * INP + 26] = 0.0f; midT[i * INP + 27] = 0.0f;
  }

  float l1acc = 0.0f, ssimacc = 0.0f;
  const float C1 = 1e-4f, C2 = 9e-4f;

  for (int d = 0; d < S + HALO; ++d) {
    const int slot = d % WIN;
    if (d < S) {
      // ---- load halo plane (zero-padded), accumulate interior L1 ----
      const float* xs = xb + (long long)d * S * S;
      const float* ys = yb + (long long)d * S * S;
      for (int idx = lane; idx < INW * INW; idx += 32) {
        int r  = idx / INW;
        int c  = idx - r * INW;
        int hh = h0 - HALO + r;
        int ww = w0 - HALO + c;
        float xv = 0.0f, yv = 0.0f;
        if ((unsigned)hh < (unsigned)S && (unsigned)ww < (unsigned)S) {
          xv = xs[hh * S + ww];
          yv = ys[hh * S + ww];
        }
        xpl[r * INP + c] = xv;
        ypl[r * INP + c] = yv;
        if (r >= HALO && r < HALO + TS && c >= HALO && c < HALO + TS)
          l1acc += fabsf(yv - xv);
      }

      // ---- stage 1: blur along W (contiguous axis) via WMMA ----
      // out26x16 in two 16-row groups (rows 0..15 and 10..25).
#pragma unroll
      for (int rg = 0; rg < 2; ++rg) {
        const int rowBase = rg * 10;
        const int row = rowBase + m;            // A-matrix M = lane&15
        v8f cX = {}, cY = {}, cXX = {}, cYY = {}, cXY = {};
#pragma unroll
        for (int j = 0; j < 7; ++j) {
          int klo = 4 * j + (hi << 1);
          v2f Xv = *(const v2f*)&xpl[row * INP + klo];
          v2f Yv = *(const v2f*)&ypl[row * INP + klo];
          v2f g  = gw[j];
          cX  = wmma4(Xv,      g, cX);
          cY  = wmma4(Yv,      g, cY);
          cXX = wmma4(Xv * Xv, g, cXX);
          cYY = wmma4(Yv * Yv, g, cYY);
          cXY = wmma4(Xv * Yv, g, cXY);
        }
        // C/D layout: element i -> (M = i + 8*hi, N = lane&15)
#pragma unroll
        for (int i2 = 0; i2 < 8; ++i2) {
          int orow = rowBase + i2 + (hi << 3);
          midT[(0 * 16 + m) * INP + orow] = cX[i2];
          midT[(1 * 16 + m) * INP + orow] = cY[i2];
          midT[(2 * 16 + m) * INP + orow] = cXX[i2];
          midT[(3 * 16 + m) * INP + orow] = cYY[i2];
          midT[(4 * 16 + m) * INP + orow] = cXY[i2];
        }
      }

      // ---- stage 2: blur along H via WMMA: out = G^T(16x28) x mid(28x16) ----
#pragma unroll
      for (int f = 0; f < NF; ++f) {
        v8f c2 = {};
#pragma unroll
        for (int j = 0; j < 7; ++j) {
          int klo = 4 * j + (hi << 1);
          v2f bb = *(const v2f*)&midT[(f * 16 + m) * INP + klo];
          c2 = wmma4(gw[j], bb, c2);
        }
#pragma unroll
        for (int i2 = 0; i2 < 8; ++i2) {
          int lin = (i2 + (hi << 3)) * 16 + m;  // (H row)*16 + (W col)
          ring[(f * WIN + slot) * PLANE + lin] = c2[i2];
        }
      }
    } else {
      // past the end of the volume: zero plane feeds the D-blur (SAME pad)
#pragma unroll
      for (int f = 0; f < NF; ++f)
#pragma unroll
        for (int j2 = 0; j2 < 8; ++j2)
          ring[(f * WIN + slot) * PLANE + lane + 32 * j2] = 0.0f;
    }

    // ---- stage 3: blur along D (11-tap on VALU) + SSIM accumulate ----
    if (d >= HALO) {
      // output depth dz = d-5; tap t reads depth q = d-10+t -> slot (d+1+t)%11
      int sl[WIN];
#pragma unroll
      for (int t2 = 0; t2 < WIN; ++t2) sl[t2] = (d + 1 + t2) % WIN;
#pragma unroll
      for (int j2 = 0; j2 < 8; ++j2) {
        int l = lane + 32 * j2;
        float a0 = 0, a1 = 0, a2 = 0, a3 = 0, a4 = 0;
#pragma unroll
        for (int t2 = 0; t2 < WIN; ++t2) {
          float w = GW[t2];
          int s2 = sl[t2];
          a0 += w * ring[(0 * WIN + s2) * PLANE + l];
          a1 += w * ring[(1 * WIN + s2) * PLANE + l];
          a2 += w * ring[(2 * WIN + s2) * PLANE + l];
          a3 += w * ring[(3 * WIN + s2) * PLANE + l];
          a4 += w * ring[(4 * WIN + s2) * PLANE + l];
        }
        float sx  = a2 - a0 * a0;
        float sy  = a3 - a1 * a1;
        float sxy = a4 - a0 * a1;
        float num = (2.0f * a0 * a1 + C1) * (2.0f * sxy + C2);
        float den = (a0 * a0 + a1 * a1 + C1) * (sx + sy + C2);
        ssimacc += num / den;
      }
    }
  }

  // wave32 reduction, lane 0 writes this tile's partials (deterministic).
#pragma unroll
  for (int off = 16; off > 0; off >>= 1) {
    l1acc   += __shfl_xor(l1acc, off, 32);
    ssimacc += __shfl_xor(ssimacc, off, 32);
  }
  if (lane == 0) {
    pl1[blockIdx.x] = l1acc;
    pss[blockIdx.x] = ssimacc;
  }
}

// ---------------------------------------------------------------------------
// 2x2x2 average pool (stride 2), batch 2, cube edge S -> S/2.
// ---------------------------------------------------------------------------
__global__ void pool2_kernel(const float* __restrict__ in,
                             float* __restrict__ out, int S) {
  const int s = S >> 1;
  const long long n = 2LL * s * s * s;
  long long i = (long long)blockIdx.x * blockDim.x + threadIdx.x;
  if (i >= n) return;
  const int ss = s * s;
  const long long vol = (long long)s * ss;
  int b = (int)(i / vol);
  long long r0 = i - (long long)b * vol;
  int z  = (int)(r0 / ss);
  int r1 = (int)(r0 - (long long)z * ss);
  int y2 = r1 / s;
  int x2 = r1 - y2 * s;
  const float* p = in + (long long)b * S * S * S +
                   (long long)(2 * z) * S * S + (2 * y2) * S + 2 * x2;
  float v = p[0] + p[1] + p[S] + p[S + 1];
  const float* q = p + (long long)S * S;
  v += q[0] + q[1] + q[S] + q[S + 1];
  out[i] = v * 0.125f;
}

// ---------------------------------------------------------------------------
// Final combine: total = sum_s w_s * (0.5*L1_s/N_s + 0.5*(1 - SSIM_s/N_s))
// ---------------------------------------------------------------------------
__global__ void msssim_final_kernel(const float* __restrict__ pl1,
                                    const float* __restrict__ pss,
                                    float* __restrict__ out, int S0) {
  const int lane = threadIdx.x;
  const float wts[3] = {4.0f / 7.0f, 2.0f / 7.0f, 1.0f / 7.0f};
  float total = 0.0f;
  int off = 0;
  int S = S0;
  for (int sc = 0; sc < 3; ++sc) {
    int T = 2 * (S / 16) * (S / 16);
    float s1 = 0.0f, s2 = 0.0f;
    for (int i = lane; i < T; i += 32) {
      s1 += pl1[off + i];
      s2 += pss[off + i];
    }
#pragma unroll
    for (int o = 16; o > 0; o >>= 1) {
      s1 += __shfl_xor(s1, o, 32);
      s2 += __shfl_xor(s2, o, 32);
    }
    float N = 2.0f * (float)S * (float)S * (float)S;
    total += wts[sc] * (0.5f * (s1 / N) + 0.5f * (1.0f - s2 / N));
    off += T;
    S >>= 1;
  }
  if (lane == 0) out[0] = total;
}

// ---------------------------------------------------------------------------
extern "C" void kernel_launch(void* const* d_in, const int* in_sizes, int n_in,
                              void* d_out, int out_size, void* d_ws,
                              size_t ws_size, hipStream_t stream) {
  (void)in_sizes; (void)n_in; (void)out_size; (void)ws_size;
  const float* real = (const float*)d_in[0];
  const float* rec  = (const float*)d_in[1];
  float* out = (float*)d_out;
  float* ws  = (float*)d_ws;

  const int S0 = 192, S1 = 96, S2 = 48;
  const int T0 = 2 * (S0 / 16) * (S0 / 16);   // 288
  const int T1 = 2 * (S1 / 16) * (S1 / 16);   // 72
  const int T2 = 2 * (S2 / 16) * (S2 / 16);   // 18

  // workspace layout (floats): partials, then pooled volumes
  float* pl1 = ws;                 // 378 used
  float* pss = ws + 512;
  float* r1  = ws + 1024;
  const long long n1 = 2LL * S1 * S1 * S1;
  float* c1 = r1 + n1;
  float* r2 = c1 + n1;
  const long long n2 = 2LL * S2 * S2 * S2;
  float* c2 = r2 + n2;

  const size_t lds =
      (size_t)(2 * INW * INP + NF * 16 * INP + NF * WIN * PLANE) * sizeof(float);
  hipFuncSetAttribute((const void*)msssim_scale_kernel,
                      hipFuncAttributeMaxDynamicSharedMemorySize, (int)lds);

  // scale 0
  msssim_scale_kernel<<<dim3(T0), dim3(32), lds, stream>>>(real, rec, S0,
                                                           pl1, pss);
  // pool 192 -> 96
  {
    int bs = 256;
    unsigned g = (unsigned)((n1 + bs - 1) / bs);
    pool2_kernel<<<dim3(g), dim3(bs), 0, stream>>>(real, r1, S0);
    pool2_kernel<<<dim3(g), dim3(bs), 0, stream>>>(rec,  c1, S0);
  }
  // scale 1
  msssim_scale_kernel<<<dim3(T1), dim3(32), lds, stream>>>(r1, c1, S1,
                                                           pl1 + T0, pss + T0);
  // pool 96 -> 48
  {
    int bs = 256;
    unsigned g = (unsigned)((n2 + bs - 1) / bs);
    pool2_kernel<<<dim3(g), dim3(bs), 0, stream>>>(r1, r2, S1);
    pool2_kernel<<<dim3(g), dim3(bs), 0, stream>>>(c1, c2, S1);
  }
  // scale 2
  msssim_scale_kernel<<<dim3(T2), dim3(32), lds, stream>>>(
      r2, c2, S2, pl1 + T0 + T1, pss + T0 + T1);

  // combine
  msssim_final_kernel<<<dim3(1), dim3(32), 0, stream>>>(pl1, pss, out, S0);
}